// SBPerformerFeatures_47304769798170
// MI455X (gfx1250) — compile-verified
//
#include <hip/hip_runtime.h>
#include <math.h>

// ---------------------------------------------------------------------------
// Performer (FAVOR+) softmax feature map for MI455X (gfx1250, wave32, WMMA).
//
// Shapes: B=4 H=16 S=4096 D=64, m=256 (proj rows = 128).
// Memory-bound (~0.67 GB traffic @ 23.3 TB/s ≈ 29 us floor) -> keep f32 and
// use V_WMMA_F32_16X16X4_F32 for the (S x 64) @ (64 x 128) projection GEMM.
//
// B fragments live in LDS feature-major with PAD=68 floats/row:
//   - fragment (k,k+1) is one aligned ds_load_b64 -> even VGPR pair, no movs
//   - PAD % 64 == 4 -> the wave's 64 loaded dwords tile all 64 banks once
// ---------------------------------------------------------------------------

typedef __attribute__((ext_vector_type(2))) float v2f;
typedef __attribute__((ext_vector_type(8))) float v8f;

#define B_      4
#define H_      16
#define S_      4096
#define D_      64
#define MHALF_  128          // m/2 (projection rows)
#define M_      256          // m
#define WAVES_PER_BLOCK 8
#define PAD     68           // LDS floats per feature row (even, ==4 mod 64)

static constexpr float kNormalizer = 0.35355339059327373f;  // D^(-1/4)
static constexpr float kDiagCoef   = 0.0625f;               // 0.5 * normalizer^2
static constexpr float kHalfLogM   = 2.772588722239781f;    // log(256)/2

#define MODE_Q      0
#define MODE_KMAX   1
#define MODE_KWRITE 2

__device__ inline void atomicMaxF(float* addr, float v) {
  // IEEE-754 ordered atomic max (values may be negative; slot init'd to -inf)
  if (v >= 0.0f) atomicMax((int*)addr, __float_as_int(v));
  else           atomicMin((unsigned int*)addr, __float_as_uint(v));
}

template <int MODE>
__global__ __launch_bounds__(256)
void performer_kernel(const float* __restrict__ X,     // (B*H*S, 64)
                      const float* __restrict__ proj,  // (128, 64)
                      float* __restrict__ feats,       // (B*H*S, 256)  (Q / KWRITE)
                      float* __restrict__ logscale,    // Q: per-row; KMAX: ws[64]
                      const float* __restrict__ ls_in) // KWRITE: ws[64] (per-bh max)
{
  // --- stage normalizer*proj into LDS, feature-major, padded row -----------
  __shared__ float psm[MHALF_ * PAD];
  for (int idx = threadIdx.x; idx < MHALF_ * D_; idx += 256) {
    int j = idx >> 6;                 // 0..127 (feature)
    int k = idx & 63;                 // 0..63  (contraction dim)
    psm[j * PAD + k] = kNormalizer * proj[idx];   // proj[j*64 + k], coalesced
  }
  __syncthreads();

  const int lane    = threadIdx.x & 31;
  const int wave    = threadIdx.x >> 5;
  const int tile    = blockIdx.x * WAVES_PER_BLOCK + wave; // 16-row tile id
  const int rowBase = tile * 16;
  const int half    = lane >> 4;      // 0: K%4 in {0,1}, 1: K%4 in {2,3}
  const int l15     = lane & 15;
  const int koff    = half * 2;

  // --- A fragments: 16 steps of 16x4 f32 (ISA 7.12.2 layout) + row sumsq ---
  const float* xrow = X + (size_t)(rowBase + l15) * D_;
  v2f   a[16];
  float ssq = 0.0f;
#pragma unroll
  for (int t = 0; t < 16; ++t) {
    a[t] = *(const v2f*)(xrow + 4 * t + koff);
    ssq += a[t].x * a[t].x + a[t].y * a[t].y;
  }
  ssq += __shfl_xor(ssq, 16);                  // full sum(x^2) for row rowBase+l15
  const float diagLane = kDiagCoef * ssq;      // diag_data for row (lane&15)

  // --- GEMM: 8 n-tiles x 16 K-steps of v_wmma_f32_16x16x4_f32 --------------
  const float* bbase = psm + (size_t)l15 * PAD + koff;
  v8f acc[8];
#pragma unroll
  for (int nt = 0; nt < 8; ++nt) {
    const float* bcol = bbase + (size_t)(nt * 16) * PAD;
    v8f c = {};
#pragma unroll
    for (int t = 0; t < 16; ++t) {
      v2f b = *(const v2f*)(bcol + 4 * t);     // one ds_load_b64, aligned pair
      c = __builtin_amdgcn_wmma_f32_16x16x4_f32(false, a[t], false, b,
                                                (short)0, c, false, false);
    }
    acc[nt] = c;
  }

  // --- per-row max|x| over all 128 projected features ----------------------
  // C/D layout: VGPR r holds row r (lanes 0-15) and row r+8 (lanes 16-31).
  float amax[8];
#pragma unroll
  for (int r = 0; r < 8; ++r) {
    float mx = 0.0f;                           // max over concat[x,-x] = max|x| >= 0
#pragma unroll
    for (int nt = 0; nt < 8; ++nt) mx = fmaxf(mx, fabsf(acc[nt][r]));
    mx = fmaxf(mx, __shfl_xor(mx, 1));
    mx = fmaxf(mx, __shfl_xor(mx, 2));
    mx = fmaxf(mx, __shfl_xor(mx, 4));
    mx = fmaxf(mx, __shfl_xor(mx, 8));
    amax[r] = mx;
  }

  float waveMax = -INFINITY;                   // KMAX accumulator
#pragma unroll
  for (int r = 0; r < 8; ++r) {
    const int   rowG  = rowBase + r + half * 8;
    const float diagR = __shfl(diagLane, r + half * 8); // row's diag (src lane < 16)
    const float am    = amax[r];

    if (MODE == MODE_Q) {
      float* f0 = feats + (size_t)rowG * M_;
#pragma unroll
      for (int nt = 0; nt < 8; ++nt) {
        const float x = acc[nt][r];
        f0[nt * 16 + l15]          = __expf(x - am);
        f0[MHALF_ + nt * 16 + l15] = __expf(-x - am);
      }
      if (l15 == 0) logscale[rowG] = am - diagR - kHalfLogM;
    } else if (MODE == MODE_KMAX) {
      waveMax = fmaxf(waveMax, am - diagR - kHalfLogM);
    } else { // MODE_KWRITE
      const float ls = ls_in[tile >> 8];       // bh = tile / (S/16)
      const float sh = diagR + kHalfLogM + ls;
      float* f0 = feats + (size_t)rowG * M_;
#pragma unroll
      for (int nt = 0; nt < 8; ++nt) {
        const float x = acc[nt][r];
        f0[nt * 16 + l15]          = __expf(x - sh);
        f0[MHALF_ + nt * 16 + l15] = __expf(-x - sh);
      }
    }
  }

  if (MODE == MODE_KMAX) {
    waveMax = fmaxf(waveMax, __shfl_xor(waveMax, 1));
    waveMax = fmaxf(waveMax, __shfl_xor(waveMax, 2));
    waveMax = fmaxf(waveMax, __shfl_xor(waveMax, 4));
    waveMax = fmaxf(waveMax, __shfl_xor(waveMax, 8));
    waveMax = fmaxf(waveMax, __shfl_xor(waveMax, 16));
    if (lane == 0) atomicMaxF(&logscale[tile >> 8], waveMax);
  }
}

__global__ void init_ws_kernel(float* ws) {
  if (threadIdx.x < B_ * H_) ws[threadIdx.x] = -INFINITY;
}

__global__ void copy_scale_kernel(const float* __restrict__ ws,
                                  float* __restrict__ out) {
  if (threadIdx.x < B_ * H_) out[threadIdx.x] = ws[threadIdx.x];
}

extern "C" void kernel_launch(void* const* d_in, const int* in_sizes, int n_in,
                              void* d_out, int out_size, void* d_ws, size_t ws_size,
                              hipStream_t stream) {
  const float* q    = (const float*)d_in[0];
  const float* k    = (const float*)d_in[1];
  const float* proj = (const float*)d_in[2];

  float* out = (float*)d_out;
  const size_t NROW = (size_t)B_ * H_ * S_;          // 262144
  float* q_feats = out;                              // NROW*256
  float* q_ls    = q_feats + NROW * (size_t)M_;      // NROW
  float* k_feats = q_ls + NROW;                      // NROW*256
  float* k_ls    = k_feats + NROW * (size_t)M_;      // B*H = 64
  float* ws      = (float*)d_ws;                     // 64 floats of scratch

  const int tiles  = (int)(NROW / 16);               // 16384 16-row tiles
  const int blocks = tiles / WAVES_PER_BLOCK;        // 2048 blocks of 8 waves

  init_ws_kernel<<<1, 64, 0, stream>>>(ws);

  performer_kernel<MODE_Q><<<blocks, 256, 0, stream>>>(
      q, proj, q_feats, q_ls, nullptr);

  performer_kernel<MODE_KMAX><<<blocks, 256, 0, stream>>>(
      k, proj, nullptr, ws, nullptr);

  copy_scale_kernel<<<1, 64, 0, stream>>>(ws, k_ls);

  performer_kernel<MODE_KWRITE><<<blocks, 256, 0, stream>>>(
      k, proj, k_feats, nullptr, ws);
}